// Net_13005160972977
// MI455X (gfx1250) — compile-verified
//
#include <hip/hip_runtime.h>
#include <hip/hip_bf16.h>
#include <cstdint>
#include <cstddef>

#define N_NODES 100000
#define N_EDGES 3200000

typedef __attribute__((ext_vector_type(2))) float v2f;
typedef __attribute__((ext_vector_type(8))) float v8f;

__device__ __forceinline__ void atomAddF(float* p, float v) {
    // relaxed, agent scope, no return -> global_atomic_add_f32 (STOREcnt path)
    __hip_atomic_fetch_add(p, v, __ATOMIC_RELAXED, __HIP_MEMORY_SCOPE_AGENT);
}

// ---------------- degree / normalization ----------------
__global__ void k_init_deg(float* __restrict__ deg) {
    int i = blockIdx.x * blockDim.x + threadIdx.x;
    if (i < N_NODES) deg[i] = 1.0f;   // self-loop contribution
}

__global__ void k_deg_accum(const long long* __restrict__ ei, float* __restrict__ deg) {
    int e = blockIdx.x * blockDim.x + threadIdx.x;
    if (e < N_EDGES) {
        int d = (int)ei[N_EDGES + e];
        atomAddF(&deg[d], 1.0f);
    }
}

__global__ void k_dinv(float* __restrict__ dinv) {
    int i = blockIdx.x * blockDim.x + threadIdx.x;
    if (i < N_NODES) dinv[i] = rsqrtf(dinv[i]);   // deg >= 1 always
}

// ---------------- layer 1: din=1 outer product, pre-scaled by dinv ----------------
__global__ void k_layer1(const float* __restrict__ x, const float* __restrict__ W1,
                         const float* __restrict__ dinv,
                         float* __restrict__ hs, float* __restrict__ acc) {
    int i = blockIdx.x * blockDim.x + threadIdx.x;
    if (i < N_NODES) {
        float s = x[i] * dinv[i];
        #pragma unroll
        for (int j = 0; j < 16; ++j) {
            float v = s * W1[j];
            hs[i * 16 + j]  = v;   // scaled message table
            acc[i * 16 + j] = v;   // accumulator starts with self-loop term
        }
    }
}

// ---------------- WMMA GEMM: hs = dinv ⊙ (X @ W), acc = hs ----------------
// One 16x16 output tile per wave; K stepped by 4 with V_WMMA_F32_16X16X4_F32.
template<int K, int WCOLS>
__global__ __launch_bounds__(128) void k_gemm_scale(const float* __restrict__ X,
                                                    const float* __restrict__ W,
                                                    const float* __restrict__ dinv,
                                                    float* __restrict__ hs,
                                                    float* __restrict__ acc) {
    constexpr int NTILES = (WCOLS + 15) / 16;
    constexpr int MTILES = N_NODES / 16;       // 100000 % 16 == 0
    constexpr int TOTAL  = MTILES * NTILES;

    int wave = blockIdx.x * (blockDim.x >> 5) + (threadIdx.x >> 5);
    if (wave >= TOTAL) return;                 // whole-wave guard: EXEC all-1 inside
    int mtile = wave / NTILES;
    int ntile = wave % NTILES;

    int lane = threadIdx.x & 31;
    int half = lane >> 4;                      // 0: lanes 0-15, 1: lanes 16-31
    int lm   = lane & 15;

    int arow = mtile * 16 + lm;                // A: row per lane (M), K split by half
    int bcol = ntile * 16 + lm;                // B: col per lane (N), K split by half

    v8f c = {};
    #pragma unroll
    for (int kk = 0; kk < K; kk += 4) {
        v2f a, b;
        a.x = X[(size_t)arow * K + kk + 2 * half];
        a.y = X[(size_t)arow * K + kk + 2 * half + 1];
        if (bcol < WCOLS) {
            b.x = W[(size_t)(kk + 2 * half)     * WCOLS + bcol];
            b.y = W[(size_t)(kk + 2 * half + 1) * WCOLS + bcol];
        } else {
            b.x = 0.0f; b.y = 0.0f;            // pad N for WCOLS=5
        }
        c = __builtin_amdgcn_wmma_f32_16x16x4_f32(
                /*neg_a=*/false, a, /*neg_b=*/false, b,
                /*c_mod=*/(short)0, c, /*reuse_a=*/false, /*reuse_b=*/false);
    }

    if (bcol < WCOLS) {
        #pragma unroll
        for (int r = 0; r < 8; ++r) {
            int row = mtile * 16 + r + 8 * half;   // C/D layout: VGPR r -> M=r / M=r+8
            float v = dinv[row] * c[r];
            size_t o = (size_t)row * WCOLS + bcol;
            hs[o]  = v;
            acc[o] = v;                            // self-loop term pre-seeded
        }
    }
}

// ---------------- edge scatter: acc[dst] += hs[src] ----------------
template<int DOUT>
__global__ void k_scatter4(const long long* __restrict__ ei,
                           const float* __restrict__ hs, float* __restrict__ acc) {
    int e = blockIdx.x * blockDim.x + threadIdx.x;
    if (e >= N_EDGES) return;
    __builtin_prefetch(ei + e + 8192, 0, 0);       // stream edge indices ahead
    int s = (int)ei[e];
    int d = (int)ei[N_EDGES + e];
    const float4* sp = (const float4*)(hs + (size_t)s * DOUT);
    float* dp = acc + (size_t)d * DOUT;
    #pragma unroll
    for (int c = 0; c < DOUT / 4; ++c) {
        float4 v = sp[c];                          // global_load_b128, L2-resident
        atomAddF(dp + 4 * c + 0, v.x);
        atomAddF(dp + 4 * c + 1, v.y);
        atomAddF(dp + 4 * c + 2, v.z);
        atomAddF(dp + 4 * c + 3, v.w);
    }
}

__global__ void k_scatter5(const long long* __restrict__ ei,
                           const float* __restrict__ hs, float* __restrict__ acc) {
    int e = blockIdx.x * blockDim.x + threadIdx.x;
    if (e >= N_EDGES) return;
    __builtin_prefetch(ei + e + 8192, 0, 0);
    int s = (int)ei[e];
    int d = (int)ei[N_EDGES + e];
    const float* sp = hs + (size_t)s * 5;
    float* dp = acc + (size_t)d * 5;
    #pragma unroll
    for (int c = 0; c < 5; ++c) atomAddF(dp + c, sp[c]);
}

// ---------------- epilogue: x' = relu(dinv[i] * acc + b) ----------------
template<int DOUT>
__global__ void k_epilogue_relu(const float* __restrict__ acc, const float* __restrict__ dinv,
                                const float* __restrict__ b, float* __restrict__ xo) {
    int idx = blockIdx.x * blockDim.x + threadIdx.x;
    if (idx < N_NODES * DOUT) {
        int i = idx / DOUT;
        int j = idx % DOUT;
        float v = dinv[i] * acc[idx] + b[j];
        xo[idx] = fmaxf(v, 0.0f);
    }
}

// ---------------- final: bias + log_softmax over 5 classes ----------------
__global__ void k_final(const float* __restrict__ acc, const float* __restrict__ dinv,
                        const float* __restrict__ b, float* __restrict__ out) {
    int i = blockIdx.x * blockDim.x + threadIdx.x;
    if (i < N_NODES) {
        float di = dinv[i];
        float v[5];
        float mx = -3.0e38f;
        #pragma unroll
        for (int j = 0; j < 5; ++j) {
            v[j] = di * acc[i * 5 + j] + b[j];
            mx = fmaxf(mx, v[j]);
        }
        float s = 0.0f;
        #pragma unroll
        for (int j = 0; j < 5; ++j) s += __expf(v[j] - mx);
        float lse = mx + __logf(s);
        #pragma unroll
        for (int j = 0; j < 5; ++j) out[i * 5 + j] = v[j] - lse;
    }
}

extern "C" void kernel_launch(void* const* d_in, const int* in_sizes, int n_in,
                              void* d_out, int out_size, void* d_ws, size_t ws_size,
                              hipStream_t stream) {
    (void)in_sizes; (void)n_in; (void)out_size; (void)ws_size;
    const float*     x  = (const float*)d_in[0];
    const long long* ei = (const long long*)d_in[1];
    const float* W1 = (const float*)d_in[2];
    const float* b1 = (const float*)d_in[3];
    const float* W2 = (const float*)d_in[4];
    const float* b2 = (const float*)d_in[5];
    const float* W3 = (const float*)d_in[6];
    const float* b3 = (const float*)d_in[7];
    const float* W4 = (const float*)d_in[8];
    const float* b4 = (const float*)d_in[9];
    float* out = (float*)d_out;

    char* ws = (char*)d_ws;
    const size_t SLOT = (size_t)N_NODES * 64 * sizeof(float);   // 25.6 MB
    float* dinv = (float*)ws;                                   // 400 KB used
    float* hs   = (float*)(ws + (size_t)512 * 1024);
    float* acc  = (float*)(ws + (size_t)512 * 1024 + SLOT);
    float* xb   = (float*)(ws + (size_t)512 * 1024 + 2 * SLOT);

    const int TB = 256;
    const int gN = (N_NODES + TB - 1) / TB;
    const int gE = (N_EDGES + TB - 1) / TB;

    // normalization: deg = 1 + in-degree; dinv = rsqrt(deg)
    k_init_deg<<<gN, TB, 0, stream>>>(dinv);
    k_deg_accum<<<gE, TB, 0, stream>>>(ei, dinv);
    k_dinv<<<gN, TB, 0, stream>>>(dinv);

    // ---- layer 1: 1 -> 16 ----
    k_layer1<<<gN, TB, 0, stream>>>(x, W1, dinv, hs, acc);
    k_scatter4<16><<<gE, TB, 0, stream>>>(ei, hs, acc);
    k_epilogue_relu<16><<<(N_NODES * 16 + TB - 1) / TB, TB, 0, stream>>>(acc, dinv, b1, xb);

    // ---- layer 2: 16 -> 64 ----
    {
        constexpr int TOT = (N_NODES / 16) * 4;                 // 25000 wave-tiles
        k_gemm_scale<16, 64><<<(TOT + 3) / 4, 128, 0, stream>>>(xb, W2, dinv, hs, acc);
    }
    k_scatter4<64><<<gE, TB, 0, stream>>>(ei, hs, acc);
    k_epilogue_relu<64><<<(N_NODES * 64 + TB - 1) / TB, TB, 0, stream>>>(acc, dinv, b2, xb);

    // ---- layer 3: 64 -> 64 ----
    {
        constexpr int TOT = (N_NODES / 16) * 4;
        k_gemm_scale<64, 64><<<(TOT + 3) / 4, 128, 0, stream>>>(xb, W3, dinv, hs, acc);
    }
    k_scatter4<64><<<gE, TB, 0, stream>>>(ei, hs, acc);
    k_epilogue_relu<64><<<(N_NODES * 64 + TB - 1) / TB, TB, 0, stream>>>(acc, dinv, b3, xb);

    // ---- layer 4: 64 -> 5 (+ log_softmax) ----
    {
        constexpr int TOT = (N_NODES / 16) * 1;                 // 6250 wave-tiles
        k_gemm_scale<64, 5><<<(TOT + 3) / 4, 128, 0, stream>>>(xb, W4, dinv, hs, acc);
    }
    k_scatter5<<<gE, TB, 0, stream>>>(ei, hs, acc);
    k_final<<<gN, TB, 0, stream>>>(acc, dinv, b4, out);
}